// DepthWiseCorr_66735201845653
// MI455X (gfx1250) — compile-verified
//
#include <hip/hip_runtime.h>
#include <stdint.h>

// ---------------------------------------------------------------------------
// Batched depthwise cross-correlation (SiamFC):
//   out[b,i,j,c] = sum_{u,v} x[b,i+u,j+v,c] * z[b,u,v,c]
// B=128, x:31x31x512, z:7x7x512, out:25x25x512, float32, NHWC (c innermost).
// Bandwidth-bound (~429 MB min traffic @ 23.3 TB/s ~= 18.4 us). Strategy:
// each x byte read from HBM exactly once via async global->LDS (CDNA5 path),
// 32x spatial reuse out of LDS, register-blocked sliding-window FMAs.
// ---------------------------------------------------------------------------

namespace {
constexpr int kB   = 128;
constexpr int kHx  = 31;
constexpr int kWx  = 31;
constexpr int kC   = 512;
constexpr int kHz  = 7;
constexpr int kWz  = 7;
constexpr int kHo  = 25;
constexpr int kWo  = 25;
constexpr int kCB  = 16;               // channels per block (64B segments)
constexpr int kCPAD = 20;              // LDS floats per spatial position (16 + pad)
                                       // 20*4B = 80B: 16B-aligned for b128 writes;
                                       // 31*20 mod 64 = 44 -> half-wave bank groups disjoint
constexpr int kNPos = kHx * kWx;       // 961
constexpr int kK    = kHz * kWz;       // 49
constexpr int kTPB  = 256;             // 8 waves (wave32)
constexpr int kNChunks = kC / kCB;     // 32
}  // namespace

// ---- CDNA5 async global->LDS copy (ASYNCcnt-tracked), with asm fallback ----
typedef int v4i __attribute__((vector_size(16)));
typedef __attribute__((address_space(1))) v4i* g_v4i_ptr;  // global (AS1)
typedef __attribute__((address_space(3))) v4i* l_v4i_ptr;  // LDS (AS3)

__device__ __forceinline__ void async_g2lds_b128(const float* __restrict__ g, float* l) {
#if __has_builtin(__builtin_amdgcn_global_load_async_to_lds_b128)
  // Low 32 bits of a flat LDS address ARE the LDS offset on gfx1250
  // (ISA 10.2); global flat == AS1 value.
  __builtin_amdgcn_global_load_async_to_lds_b128(
      (g_v4i_ptr)(uintptr_t)g,
      (l_v4i_ptr)(unsigned)(uintptr_t)l,
      /*offset=*/0, /*cpol=*/0);
#else
  asm volatile("global_load_async_to_lds_b128 %0, %1, off"
               :
               : "v"((unsigned)(uintptr_t)l),
                 "v"((unsigned long long)(uintptr_t)g)
               : "memory");
#endif
}

__device__ __forceinline__ void wait_async_lds() {
#if __has_builtin(__builtin_amdgcn_s_wait_asynccnt)
  __builtin_amdgcn_s_wait_asynccnt(0);
#else
  asm volatile("s_wait_asynccnt 0" ::: "memory");
#endif
}

__global__ __launch_bounds__(kTPB) void DepthWiseCorr_66735201845653_kernel(
    const float* __restrict__ x, const float* __restrict__ z,
    float* __restrict__ out) {
  __shared__ float lx[kNPos * kCPAD];  // 961*20*4 = 76,880 B
  __shared__ float lz[kK * kCB];       // 49*16*4  =  3,136 B

  const int blk = blockIdx.x;
  const int b   = blk >> 5;            // / kNChunks (32)
  const int c0  = (blk & (kNChunks - 1)) * kCB;
  const int tid = threadIdx.x;

  const float* xg = x + ((size_t)b * kNPos) * kC + c0;
  const float* zg = z + ((size_t)b * kK) * kC + c0;

  // ---- Stage x tile: 961 positions x 16 floats = 3844 float4 tasks --------
  for (int task = tid; task < kNPos * 4; task += kTPB) {
    const int p = task >> 2;
    const int q = task & 3;
    async_g2lds_b128(xg + (size_t)p * kC + q * 4, &lx[p * kCPAD + q * 4]);
  }
  // ---- Stage z tile: 49 taps x 16 floats = 196 float4 tasks ---------------
  for (int task = tid; task < kK * 4; task += kTPB) {
    const int k = task >> 2;
    const int q = task & 3;
    async_g2lds_b128(zg + (size_t)k * kC + q * 4, &lz[k * kCB + q * 4]);
  }
  wait_async_lds();
  __syncthreads();

  // ---- Per-thread z taps in registers (c fixed across this thread's units)
  const int c = tid & (kCB - 1);
  float zr[kK];
#pragma unroll
  for (int k = 0; k < kK; ++k) zr[k] = lz[k * kCB + c];

  // ---- Compute: work unit = (output row i, channel c); 25*16 = 400 units --
  for (int unit = tid; unit < kHo * kCB; unit += kTPB) {
    const int i = unit >> 4;  // output row (unit & 15 == c, invariant)

    float acc[kWo];
#pragma unroll
    for (int j = 0; j < kWo; ++j) acc[j] = 0.0f;

#pragma unroll
    for (int u = 0; u < kHz; ++u) {
      // one 31-float x row for this channel, sliding-window reused 7x
      float xr[kWx];
      const int pbase = (i + u) * kWx;
#pragma unroll
      for (int w = 0; w < kWx; ++w) xr[w] = lx[(pbase + w) * kCPAD + c];
#pragma unroll
      for (int v = 0; v < kWz; ++v) {
        const float zv = zr[u * kWz + v];
#pragma unroll
        for (int j = 0; j < kWo; ++j) acc[j] = __builtin_fmaf(xr[j + v], zv, acc[j]);
      }
    }

    float* og = out + (((size_t)b * kHo + i) * kWo) * kC + c0 + c;
#pragma unroll
    for (int j = 0; j < kWo; ++j) og[(size_t)j * kC] = acc[j];
  }
}

extern "C" void kernel_launch(void* const* d_in, const int* in_sizes, int n_in,
                              void* d_out, int out_size, void* d_ws, size_t ws_size,
                              hipStream_t stream) {
  (void)in_sizes; (void)n_in; (void)out_size; (void)d_ws; (void)ws_size;
  const float* x = (const float*)d_in[0];   // [128,31,31,512] f32
  const float* z = (const float*)d_in[1];   // [128,7,7,512]   f32
  float* out = (float*)d_out;               // [128,25,25,512] f32

  dim3 grid(kB * kNChunks);                 // 4096 blocks
  DepthWiseCorr_66735201845653_kernel<<<grid, kTPB, 0, stream>>>(x, z, out);
}